// LDRLD_84653805404807
// MI455X (gfx1250) — compile-verified
//
#include <hip/hip_runtime.h>
#include <math.h>

typedef __attribute__((ext_vector_type(4))) float v4f;
typedef __attribute__((ext_vector_type(2))) float v2f;
typedef __attribute__((ext_vector_type(8))) float v8f;

#define VOCAB    32000
#define NROWS    2048
#define TEMP_F   4.0f
#define INV_T    0.25f
#define NTHREADS 256
#define NWAVES   8

__device__ __forceinline__ float wexp(float x) { return __expf(x); }
__device__ __forceinline__ float wlog(float x) { return __logf(x); }

// KL over a restricted 2-element softmax: a = teacher/T logits, b = student/T logits
__device__ __forceinline__ float kl2(float a0, float a1, float b0, float b1) {
    float ma = fmaxf(a0, a1);
    float e0 = wexp(a0 - ma), e1 = wexp(a1 - ma);
    float Za = e0 + e1;
    float A  = ma + wlog(Za);
    float mb = fmaxf(b0, b1);
    float Zb = wexp(b0 - mb) + wexp(b1 - mb);
    float Bv = mb + wlog(Zb);
    float inv = 1.0f / Za;
    return (e0 * ((a0 - A) - (b0 - Bv)) + e1 * ((a1 - A) - (b1 - Bv))) * inv;
}

__device__ __forceinline__ float kl3(float a0, float a1, float a2,
                                     float b0, float b1, float b2) {
    float ma = fmaxf(a0, fmaxf(a1, a2));
    float e0 = wexp(a0 - ma), e1 = wexp(a1 - ma), e2 = wexp(a2 - ma);
    float Za = e0 + e1 + e2;
    float A  = ma + wlog(Za);
    float mb = fmaxf(b0, fmaxf(b1, b2));
    float Zb = wexp(b0 - mb) + wexp(b1 - mb) + wexp(b2 - mb);
    float Bv = mb + wlog(Zb);
    float inv = 1.0f / Za;
    return (e0 * ((a0 - A) - (b0 - Bv)) +
            e1 * ((a1 - A) - (b1 - Bv)) +
            e2 * ((a2 - A) - (b2 - Bv))) * inv;
}

__global__ __launch_bounds__(NTHREADS)
void row_loss_kernel(const float* __restrict__ S, const float* __restrict__ T,
                     const int* __restrict__ tgt, float* __restrict__ ws)
{
    const int row = blockIdx.x;
    const size_t base = (size_t)row * VOCAB;
    const v4f* S4 = (const v4f*)(S + base);
    const v4f* T4 = (const v4f*)(T + base);
    const int tid = threadIdx.x;

    // Online accumulators.
    // Student: shared max m_s; s1 = sum exp(s - m_s) (temp 1), sb = sum exp((s-m_s)/T)
    float m_s = -__builtin_inff(), s1 = 0.0f, sb = 0.0f;
    // Teacher: m_t max; sa = sum exp((t-m_t)/T); w = sum exp((t-m_t)/T)*(t-s)/T
    float m_t = -__builtin_inff(), sa = 0.0f, w = 0.0f;
    // Top-3 of student (desc): values v*, indices i*, teacher values at same index t*
    float v0 = -__builtin_inff(), v1 = -__builtin_inff(), v2 = -__builtin_inff();
    int   i0 = -1, i1 = -1, i2 = -1;
    float t0 = 0.0f, t1 = 0.0f, t2 = 0.0f;

    for (int i = tid; i < VOCAB / 4; i += NTHREADS) {
        v4f s4 = __builtin_nontemporal_load(S4 + i);   // global_load_b128 th:NT
        v4f t4 = __builtin_nontemporal_load(T4 + i);
        #pragma unroll
        for (int c = 0; c < 4; ++c) {
            float s = s4[c], t = t4[c];
            // student (max shared between temp-1 and temp-T sums)
            if (s > m_s) {
                float d = m_s - s;
                s1 *= wexp(d);
                sb *= wexp(d * INV_T);
                m_s = s;
            }
            float es = s - m_s;
            s1 += wexp(es);
            sb += wexp(es * INV_T);
            // teacher
            if (t > m_t) {
                float cs = wexp((m_t - t) * INV_T);
                sa *= cs; w *= cs;
                m_t = t;
            }
            float ea = wexp((t - m_t) * INV_T);
            sa += ea;
            w  += ea * (t - s) * INV_T;
            // running top-3 of student
            if (s > v2) {
                int j = i * 4 + c;
                if (s > v0)      { v2=v1; i2=i1; t2=t1; v1=v0; i1=i0; t1=t0; v0=s; i0=j; t0=t; }
                else if (s > v1) { v2=v1; i2=i1; t2=t1; v1=s;  i1=j;  t1=t; }
                else             { v2=s;  i2=j;  t2=t; }
            }
        }
    }

    // ---- wave32 butterfly reduction ----
    #pragma unroll
    for (int off = 16; off > 0; off >>= 1) {
        float om  = __shfl_xor(m_s, off);
        float os1 = __shfl_xor(s1, off);
        float osb = __shfl_xor(sb, off);
        float nm  = fmaxf(m_s, om);
        s1 = s1 * wexp(m_s - nm) + os1 * wexp(om - nm);
        sb = sb * wexp((m_s - nm) * INV_T) + osb * wexp((om - nm) * INV_T);
        m_s = nm;

        float omt = __shfl_xor(m_t, off);
        float osa = __shfl_xor(sa, off);
        float ow  = __shfl_xor(w, off);
        float nmt = fmaxf(m_t, omt);
        float ca  = wexp((m_t - nmt) * INV_T);
        float cb  = wexp((omt - nmt) * INV_T);
        sa = sa * ca + osa * cb;
        w  = w  * ca + ow  * cb;
        m_t = nmt;

        float ov0 = __shfl_xor(v0, off), ov1 = __shfl_xor(v1, off), ov2 = __shfl_xor(v2, off);
        int   oi0 = __shfl_xor(i0, off), oi1 = __shfl_xor(i1, off), oi2 = __shfl_xor(i2, off);
        float ot0 = __shfl_xor(t0, off), ot1 = __shfl_xor(t1, off), ot2 = __shfl_xor(t2, off);
        float av[3] = {v0, v1, v2},   bv[3] = {ov0, ov1, ov2};
        int   ai[3] = {i0, i1, i2},   bi[3] = {oi0, oi1, oi2};
        float at[3] = {t0, t1, t2},   bt[3] = {ot0, ot1, ot2};
        float rv[3]; int ri[3]; float rt[3];
        int p = 0, q = 0;
        #pragma unroll
        for (int k = 0; k < 3; ++k) {
            if (av[p] >= bv[q]) { rv[k]=av[p]; ri[k]=ai[p]; rt[k]=at[p]; ++p; }
            else                { rv[k]=bv[q]; ri[k]=bi[q]; rt[k]=bt[q]; ++q; }
        }
        v0=rv[0]; v1=rv[1]; v2=rv[2];
        i0=ri[0]; i1=ri[1]; i2=ri[2];
        t0=rt[0]; t1=rt[1]; t2=rt[2];
    }

    // ---- cross-wave combine through LDS ----
    __shared__ float smf[NWAVES][12];
    __shared__ int   smi[NWAVES][3];
    const int wave = tid >> 5;
    const int lane = tid & 31;
    if (lane == 0) {
        float* f = smf[wave];
        f[0]=m_s; f[1]=s1; f[2]=sb; f[3]=m_t; f[4]=sa; f[5]=w;
        f[6]=v0;  f[7]=v1; f[8]=v2; f[9]=t0;  f[10]=t1; f[11]=t2;
        smi[wave][0]=i0; smi[wave][1]=i1; smi[wave][2]=i2;
    }
    __syncthreads();

    if (tid == 0) {
        for (int wv = 1; wv < NWAVES; ++wv) {
            const float* f = smf[wv];
            float om = f[0], os1 = f[1], osb = f[2];
            float nm = fmaxf(m_s, om);
            s1 = s1 * wexp(m_s - nm) + os1 * wexp(om - nm);
            sb = sb * wexp((m_s - nm) * INV_T) + osb * wexp((om - nm) * INV_T);
            m_s = nm;

            float omt = f[3], osa = f[4], ow = f[5];
            float nmt = fmaxf(m_t, omt);
            float ca = wexp((m_t - nmt) * INV_T);
            float cb = wexp((omt - nmt) * INV_T);
            sa = sa * ca + osa * cb;
            w  = w  * ca + ow  * cb;
            m_t = nmt;

            float av[3] = {v0, v1, v2},  bv[3] = {f[6], f[7], f[8]};
            int   ai[3] = {i0, i1, i2},  bi[3] = {smi[wv][0], smi[wv][1], smi[wv][2]};
            float at[3] = {t0, t1, t2},  bt[3] = {f[9], f[10], f[11]};
            float rv[3]; int ri[3]; float rt[3];
            int p = 0, q = 0;
            for (int k = 0; k < 3; ++k) {
                if (av[p] >= bv[q]) { rv[k]=av[p]; ri[k]=ai[p]; rt[k]=at[p]; ++p; }
                else                { rv[k]=bv[q]; ri[k]=bi[q]; rt[k]=bt[q]; ++q; }
            }
            v0=rv[0]; v1=rv[1]; v2=rv[2];
            i0=ri[0]; i1=ri[1]; i2=ri[2];
            t0=rt[0]; t1=rt[1]; t2=rt[2];
        }

        // ---- finalize row ----
        const float s_tgt = S[base + (size_t)tgt[row]];
        const float logZ  = m_s + wlog(s1);
        const float ce    = logZ - s_tgt;               // = -logq_full[target]

        // combos over student's top-3 (the -1000 mask underflows all other terms)
        const float a0 = t0 * INV_T, a1 = t1 * INV_T, a2 = t2 * INV_T;
        const float b0 = v0 * INV_T, b1 = v1 * INV_T, b2 = v2 * INV_T;
        const float w01 = (2.0f / 2.5f) * wexp(-0.05f * 3.0f);
        const float w02 = (2.0f / 3.5f) * wexp(-0.05f * 4.0f);
        const float w12 = (2.0f / 2.5f) * wexp(-0.05f * 5.0f);
        float kd = w01 * kl2(a0, a1, b0, b1)
                 + w02 * kl2(a0, a2, b0, b2)
                 + w12 * kl2(a1, a2, b1, b2)
                 +        kl3(a0, a1, a2, b0, b1, b2);
        const float loss_kd = kd * (TEMP_F * TEMP_F) / (float)NROWS / 4.0f;

        // "not" KL: full-row sums minus top-3 contributions
        const float ex0 = wexp((t0 - m_t) * INV_T);
        const float ex1 = wexp((t1 - m_t) * INV_T);
        const float ex2 = wexp((t2 - m_t) * INV_T);
        const float sa_ex = sa - (ex0 + ex1 + ex2);
        const float w_ex  = w  - (ex0 * (t0 - v0) + ex1 * (t1 - v1) + ex2 * (t2 - v2)) * INV_T;
        const float sb_ex = sb - (wexp((v0 - m_s) * INV_T) +
                                  wexp((v1 - m_s) * INV_T) +
                                  wexp((v2 - m_s) * INV_T));
        const float A_lse = m_t * INV_T + wlog(sa_ex);
        const float B_lse = m_s * INV_T + wlog(sb_ex);
        const float kl_not = w_ex / sa_ex - (A_lse - B_lse);
        const float loss_not = kl_not * (TEMP_F * TEMP_F) / (float)NROWS;

        ws[row] = ce / (float)NROWS + loss_kd + loss_not;   // CE_W=ALPHA=BETA=1
    }
}

// Final deterministic 2048 -> 1 reduction using V_WMMA_F32_16X16X4_F32.
// A = all-ones 16x4 so D[m][n] = sum_k B[k][n]; accumulating 32 chunks of 64
// partials into C and summing every D element counts the total exactly 16x.
__global__ __launch_bounds__(32)
void final_reduce_wmma(const float* __restrict__ ws, float* __restrict__ out)
{
    const int lane = threadIdx.x;
    v8f c = {0.f, 0.f, 0.f, 0.f, 0.f, 0.f, 0.f, 0.f};
    v2f a; a[0] = 1.0f; a[1] = 1.0f;
    for (int basei = 0; basei < NROWS; basei += 64) {
        v2f b;
        b[0] = ws[basei + 2 * lane];
        b[1] = ws[basei + 2 * lane + 1];
        c = __builtin_amdgcn_wmma_f32_16x16x4_f32(
                /*neg_a=*/false, a, /*neg_b=*/false, b,
                /*c_mod=*/(short)0, c, /*reuse_a=*/false, /*reuse_b=*/false);
    }
    float sum = 0.0f;
    #pragma unroll
    for (int k = 0; k < 8; ++k) sum += c[k];
    #pragma unroll
    for (int off = 16; off > 0; off >>= 1) sum += __shfl_xor(sum, off);
    if (lane == 0) out[0] = sum * (1.0f / 16.0f);
}

extern "C" void kernel_launch(void* const* d_in, const int* in_sizes, int n_in,
                              void* d_out, int out_size, void* d_ws, size_t ws_size,
                              hipStream_t stream) {
    const float* S   = (const float*)d_in[0];   // logits_student (B,V) f32
    const float* T   = (const float*)d_in[1];   // logits_teacher (B,V) f32
    const int*   tgt = (const int*)d_in[2];     // target (B,)
    float* ws  = (float*)d_ws;                  // 2048 per-row partials
    float* out = (float*)d_out;                 // scalar

    row_loss_kernel<<<NROWS, NTHREADS, 0, stream>>>(S, T, tgt, ws);
    final_reduce_wmma<<<1, 32, 0, stream>>>(ws, out);
}